// EdgeClassifier_68066641707575
// MI455X (gfx1250) — compile-verified
//
#include <hip/hip_runtime.h>
#include <hip/hip_bf16.h>
#include <math.h>

#define N_NODES 100000
#define N_EDGES 1600000
#define HDIM    64

typedef float v2f __attribute__((ext_vector_type(2)));
typedef float v8f __attribute__((ext_vector_type(8)));

__device__ __forceinline__ v8f wmma_f32(v2f a, v2f b, v8f c) {
  // D = A(16x4 f32) * B(4x16 f32) + C(16x16 f32)
  return __builtin_amdgcn_wmma_f32_16x16x4_f32(
      /*neg_a=*/false, a, /*neg_b=*/false, b,
      /*c_mod=*/(short)0, c, /*reuse_a=*/false, /*reuse_b=*/false);
}

// ---------------------------------------------------------------- utilities
__global__ void zero_f32(float* __restrict__ p, long n) {
  long t = (long)blockIdx.x * blockDim.x + threadIdx.x;
  if (t < n) p[t] = 0.0f;
}

__global__ void degree_kernel(const int* __restrict__ dst, float* __restrict__ deg) {
  int e = blockIdx.x * blockDim.x + threadIdx.x;
  if (e < N_EDGES) atomicAdd(&deg[dst[e]], 1.0f);
}

__global__ void invdeg_kernel(float* __restrict__ deg) {
  int i = blockIdx.x * blockDim.x + threadIdx.x;
  if (i < N_NODES) deg[i] = 1.0f / fmaxf(deg[i], 1.0f);
}

// Pack 64x64 weight into row-pair float2 layout: Wp[(k/2)*64+col] = {W[k][col], W[k+1][col]}
__global__ void pack_w64(const float* __restrict__ W, v2f* __restrict__ Wp) {
  int t = blockIdx.x * blockDim.x + threadIdx.x;
  if (t >= 32 * 64) return;
  int p = t >> 6, col = t & 63;
  v2f v;
  v.x = W[(2 * p) * HDIM + col];
  v.y = W[(2 * p + 1) * HDIM + col];
  Wp[t] = v;
}

// Pack [We1 | Wl1] (133 rows, zero-padded to 136) x 128 cols into row-pair float2s.
__global__ void pack_wedge(const float* __restrict__ We1, const float* __restrict__ Wl1,
                           v2f* __restrict__ Wp) {
  int t = blockIdx.x * blockDim.x + threadIdx.x;
  if (t >= 68 * 128) return;
  int p = t >> 7, col = t & 127;
  int k0 = 2 * p, k1 = 2 * p + 1;
  float x = 0.f, y = 0.f;
  if (k0 < 133) x = (col < 64) ? We1[k0 * HDIM + col] : Wl1[k0 * HDIM + col - 64];
  if (k1 < 133) y = (col < 64) ? We1[k1 * HDIM + col] : Wl1[k1 * HDIM + col - 64];
  v2f v; v.x = x; v.y = y;
  Wp[t] = v;
}

// SAGE layer 1 projections (K = F_NODE = 4, tiny -> plain VALU)
__global__ void sage1_proj(const float* __restrict__ nf,
                           const float* __restrict__ W1n,
                           const float* __restrict__ W1s,
                           float* __restrict__ P1, float* __restrict__ S1) {
  int t = blockIdx.x * blockDim.x + threadIdx.x;
  if (t >= N_NODES * HDIM) return;
  int i = t >> 6, j = t & 63;
  const float* f = nf + (size_t)i * 4;
  float p = 0.f, s = 0.f;
#pragma unroll
  for (int k = 0; k < 4; ++k) {
    float fk = f[k];
    p += fk * W1n[k * HDIM + j];
    s += fk * W1s[k * HDIM + j];
  }
  P1[t] = p;
  S1[t] = s;
}

// agg[dst] += P[src] (per-feature float atomics; node table is L2 resident)
__global__ void scatter_add(const float* __restrict__ P,
                            const int* __restrict__ src,
                            const int* __restrict__ dst,
                            float* __restrict__ agg) {
  long t = (long)blockIdx.x * blockDim.x + threadIdx.x;
  if (t >= (long)N_EDGES * HDIM) return;
  int e = (int)(t >> 6), j = (int)(t & 63);
  atomicAdd(&agg[(size_t)dst[e] * HDIM + j], P[(size_t)src[e] * HDIM + j]);
}

// h = relu(S + agg * inv_deg + b)  (in-place on S allowed)
__global__ void combine_relu(const float* __restrict__ S,
                             const float* __restrict__ agg,
                             const float* __restrict__ invdeg,
                             const float* __restrict__ b,
                             float* __restrict__ h) {
  int t = blockIdx.x * blockDim.x + threadIdx.x;
  if (t >= N_NODES * HDIM) return;
  int i = t >> 6, j = t & 63;
  h[t] = fmaxf(S[t] + agg[t] * invdeg[i] + b[j], 0.0f);
}

// ------------------------------------------------ WMMA GEMM: C[N x 64] = A[N x 64] @ W[64 x 64]
// One wave32 per 16-row tile; packed B -> one b64 load per fragment.
template <bool BIAS, bool RELU>
__global__ void gemm_n64(const float* __restrict__ A, const v2f* __restrict__ Wp,
                         const float* __restrict__ bias, float* __restrict__ C,
                         int nrows) {
  int wave = blockIdx.x * (blockDim.x >> 5) + (threadIdx.x >> 5);
  int lane = threadIdx.x & 31;
  int row0 = wave << 4;
  if (row0 >= nrows) return;

  int m    = lane & 15;
  int half = lane >> 4;
  int kh   = half << 1;

  const float* arow = A + (size_t)(row0 + m) * HDIM;
  v8f acc[4] = {};

#pragma unroll 4
  for (int k = 0; k < HDIM; k += 4) {
    int pair = (k >> 1) + half;          // rows (k+kh, k+kh+1)
    v2f a;
    a.x = arow[k + kh];
    a.y = arow[k + kh + 1];
#pragma unroll
    for (int t = 0; t < 4; ++t) {
      v2f b = Wp[pair * 64 + t * 16 + m];
      acc[t] = wmma_f32(a, b, acc[t]);
    }
  }

#pragma unroll
  for (int t = 0; t < 4; ++t) {
    int col = t * 16 + m;
    float bv = BIAS ? bias[col] : 0.0f;
#pragma unroll
    for (int r = 0; r < 8; ++r) {
      int row = row0 + r + 8 * half;     // C layout: lane<16 -> M=r, lane>=16 -> M=8+r
      float v = acc[t][r] + bv;
      if (RELU) v = fmaxf(v, 0.0f);
      C[(size_t)row * HDIM + col] = v;
    }
  }
}

// ------------------------------------------------ fused edge head
// One wave32 = 32 edges (two 16-row A tiles sharing every B fragment),
// N split into two 64-col passes so each pass holds only 8 v8f accumulators
// (no scratch spills). Results staged in LDS, 1-edge-per-lane epilogue.
#define EH_WAVES 2

__device__ __forceinline__ void eh_gemm_pass(
    const v2f* __restrict__ Wp, int colbase,
    const float* __restrict__ qsA, const float* __restrict__ qdA,
    const float* __restrict__ efA,
    const float* __restrict__ qsB, const float* __restrict__ qdB,
    const float* __restrict__ efB,
    int m, int half, int kh, float (*sm)[132]) {
  v8f accA[4] = {};
  v8f accB[4] = {};

  // region 0: K in [0,64) -> A from Q[src]
#pragma unroll 2
  for (int k = 0; k < 64; k += 4) {
    int ka = k + kh, pair = (k >> 1) + half;
    v2f b[4];
#pragma unroll
    for (int t = 0; t < 4; ++t) b[t] = Wp[pair * 128 + colbase + t * 16 + m];
    v2f a0 = {qsA[ka], qsA[ka + 1]};
    v2f a1 = {qsB[ka], qsB[ka + 1]};
#pragma unroll
    for (int t = 0; t < 4; ++t) {
      accA[t] = wmma_f32(a0, b[t], accA[t]);
      accB[t] = wmma_f32(a1, b[t], accB[t]);
    }
  }

  // region 1: K in [64,128) -> A from Q[dst]
#pragma unroll 2
  for (int k = 64; k < 128; k += 4) {
    int ka = k + kh, pair = (k >> 1) + half;
    v2f b[4];
#pragma unroll
    for (int t = 0; t < 4; ++t) b[t] = Wp[pair * 128 + colbase + t * 16 + m];
    v2f a0 = {qdA[ka - 64], qdA[ka - 63]};
    v2f a1 = {qdB[ka - 64], qdB[ka - 63]};
#pragma unroll
    for (int t = 0; t < 4; ++t) {
      accA[t] = wmma_f32(a0, b[t], accA[t]);
      accB[t] = wmma_f32(a1, b[t], accB[t]);
    }
  }

  // region 2a: K = 128..131 -> A from edge_feat[0..3]
  {
    int pair = 64 + half;
    v2f b[4];
#pragma unroll
    for (int t = 0; t < 4; ++t) b[t] = Wp[pair * 128 + colbase + t * 16 + m];
    v2f a0 = {efA[kh], efA[kh + 1]};
    v2f a1 = {efB[kh], efB[kh + 1]};
#pragma unroll
    for (int t = 0; t < 4; ++t) {
      accA[t] = wmma_f32(a0, b[t], accA[t]);
      accB[t] = wmma_f32(a1, b[t], accB[t]);
    }
  }
  // region 2b: K = 132..135 (rows 133-135 are zero in packed B)
  {
    bool lo = (half == 0);
    int pair = 66 + half;
    v2f b[4];
#pragma unroll
    for (int t = 0; t < 4; ++t) b[t] = Wp[pair * 128 + colbase + t * 16 + m];
    v2f a0 = {lo ? efA[4] : 0.0f, 0.0f};
    v2f a1 = {lo ? efB[4] : 0.0f, 0.0f};
#pragma unroll
    for (int t = 0; t < 4; ++t) {
      accA[t] = wmma_f32(a0, b[t], accA[t]);
      accB[t] = wmma_f32(a1, b[t], accB[t]);
    }
  }

  // spill this 64-col half to LDS (rows 0-15: tile A edges, 16-31: tile B)
#pragma unroll
  for (int t = 0; t < 4; ++t)
#pragma unroll
    for (int r = 0; r < 8; ++r) {
      int col = colbase + t * 16 + m;
      sm[r + 8 * half][col]      = accA[t][r];
      sm[16 + r + 8 * half][col] = accB[t][r];
    }
}

__global__ __launch_bounds__(EH_WAVES * 32) void edge_head(
    const float* __restrict__ Q,
    const float* __restrict__ edge_feat,
    const int* __restrict__ src, const int* __restrict__ dst,
    const v2f* __restrict__ Wp,      // packed [68][128]
    const float* __restrict__ be1, const float* __restrict__ We2,
    const float* __restrict__ be2,
    const float* __restrict__ bl1, const float* __restrict__ Wl2,
    const float* __restrict__ bl2,
    float* __restrict__ out) {
  __shared__ float sT[EH_WAVES][32][132];   // 32 edges x 128 cols (+pad)

  int wslot = threadIdx.x >> 5;
  int wave  = blockIdx.x * EH_WAVES + wslot;
  int lane  = threadIdx.x & 31;
  int e0    = wave << 5;                    // 32 edges per wave
  if (e0 >= N_EDGES) return;                // grid exact: never taken

  int m    = lane & 15;
  int half = lane >> 4;
  int kh   = half << 1;

  int eA = e0 + m, eB = e0 + 16 + m;
  const float* qsA = Q + (size_t)src[eA] * HDIM;
  const float* qdA = Q + (size_t)dst[eA] * HDIM;
  const float* qsB = Q + (size_t)src[eB] * HDIM;
  const float* qdB = Q + (size_t)dst[eB] * HDIM;
  const float* efA = edge_feat + (size_t)eA * 5;
  const float* efB = edge_feat + (size_t)eB * 5;

  eh_gemm_pass(Wp, 0,  qsA, qdA, efA, qsB, qdB, efB, m, half, kh, sT[wslot]); // T half
  eh_gemm_pass(Wp, 64, qsA, qdA, efA, qsB, qdB, efB, m, half, kh, sT[wslot]); // U half

  asm volatile("s_wait_dscnt 0" ::: "memory");

  // epilogue: one edge per lane (32 edges, 32 lanes)
  {
    const float* Trow = sT[wslot][lane];
    float be2v = be2[0], bl2v = bl2[0];

    float a1 = 0.0f;
#pragma unroll 4
    for (int j = 0; j < HDIM; ++j)
      a1 += tanhf(Trow[j] + be1[j]) * We2[j];
    float w = 1.0f / (1.0f + expf(-(a1 + be2v)));   // sigmoid

    float a2 = 0.0f;
#pragma unroll 4
    for (int j = 0; j < HDIM; ++j) {
      float u = fmaxf(w * Trow[64 + j] + bl1[j], 0.0f);
      a2 += u * Wl2[j];
    }
    out[e0 + lane] = a2 + bl2v;
  }
}

// ---------------------------------------------------------------- launcher
extern "C" void kernel_launch(void* const* d_in, const int* in_sizes, int n_in,
                              void* d_out, int out_size, void* d_ws, size_t ws_size,
                              hipStream_t stream) {
  const float* node_feat = (const float*)d_in[0];
  const float* edge_feat = (const float*)d_in[1];
  const int*   src       = (const int*)d_in[2];
  const int*   dst       = (const int*)d_in[3];
  const float* W1s = (const float*)d_in[4];
  const float* W1n = (const float*)d_in[5];
  const float* b1  = (const float*)d_in[6];
  const float* W2s = (const float*)d_in[7];
  const float* W2n = (const float*)d_in[8];
  const float* b2  = (const float*)d_in[9];
  const float* Wnp = (const float*)d_in[10];
  const float* bnp = (const float*)d_in[11];
  const float* We1 = (const float*)d_in[12];
  const float* be1 = (const float*)d_in[13];
  const float* We2 = (const float*)d_in[14];
  const float* be2 = (const float*)d_in[15];
  const float* Wl1 = (const float*)d_in[16];
  const float* bl1 = (const float*)d_in[17];
  const float* Wl2 = (const float*)d_in[18];
  const float* bl2 = (const float*)d_in[19];
  float* out = (float*)d_out;

  float* ws   = (float*)d_ws;
  const size_t NH = (size_t)N_NODES * HDIM;
  float* deg  = ws;                 // [N]
  float* bufA = deg  + N_NODES;     // [N*H]  P1 / P2 / Q
  float* bufB = bufA + NH;          // [N*H]  S1 / h1
  float* bufC = bufB + NH;          // [N*H]  S2 / h2
  float* agg  = bufC + NH;          // [N*H]
  v2f*  wp2n  = (v2f*)(agg + NH);   // [32*64]
  v2f*  wp2s  = wp2n + 32 * 64;
  v2f*  wpnp  = wp2s + 32 * 64;
  v2f*  wpe   = wpnp + 32 * 64;     // [68*128]

  const int T = 256;
  const long NHl = (long)NH;
  const long EH  = (long)N_EDGES * HDIM;

  // weight repacks (tiny, one-shot)
  pack_w64<<<(32 * 64 + T - 1) / T, T, 0, stream>>>(W2n, wp2n);
  pack_w64<<<(32 * 64 + T - 1) / T, T, 0, stream>>>(W2s, wp2s);
  pack_w64<<<(32 * 64 + T - 1) / T, T, 0, stream>>>(Wnp, wpnp);
  pack_wedge<<<(68 * 128 + T - 1) / T, T, 0, stream>>>(We1, Wl1, wpe);

  // degrees -> inverse degrees
  zero_f32<<<(N_NODES + T - 1) / T, T, 0, stream>>>(deg, N_NODES);
  degree_kernel<<<(N_EDGES + T - 1) / T, T, 0, stream>>>(dst, deg);
  invdeg_kernel<<<(N_NODES + T - 1) / T, T, 0, stream>>>(deg);

  // SAGE layer 1
  sage1_proj<<<(int)((NHl + T - 1) / T), T, 0, stream>>>(node_feat, W1n, W1s, bufA, bufB);
  zero_f32<<<(int)((NHl + T - 1) / T), T, 0, stream>>>(agg, NHl);
  scatter_add<<<(int)((EH + T - 1) / T), T, 0, stream>>>(bufA, src, dst, agg);
  combine_relu<<<(int)((NHl + T - 1) / T), T, 0, stream>>>(bufB, agg, deg, b1, bufB); // h1

  // SAGE layer 2 (WMMA GEMMs)
  const int tiles = N_NODES / 16;                     // 6250
  const int gblk  = (tiles + 7) / 8;                  // 8 waves / 256-thread block
  gemm_n64<false, false><<<gblk, 256, 0, stream>>>(bufB, wp2n, nullptr, bufA, N_NODES); // P2
  gemm_n64<false, false><<<gblk, 256, 0, stream>>>(bufB, wp2s, nullptr, bufC, N_NODES); // S2
  zero_f32<<<(int)((NHl + T - 1) / T), T, 0, stream>>>(agg, NHl);
  scatter_add<<<(int)((EH + T - 1) / T), T, 0, stream>>>(bufA, src, dst, agg);
  combine_relu<<<(int)((NHl + T - 1) / T), T, 0, stream>>>(bufC, agg, deg, b2, bufC); // h2

  // Per-node Q = relu(h2 @ Wnp + bnp)
  gemm_n64<true, true><<<gblk, 256, 0, stream>>>(bufC, wpnp, bnp, bufA, N_NODES);     // Q

  // Fused edge head: 32 edges per wave, two 64-col passes
  const int ewaves = N_EDGES / 32;                    // 50000
  edge_head<<<ewaves / EH_WAVES, EH_WAVES * 32, 0, stream>>>(
      bufA, edge_feat, src, dst, wpe, be1, We2, be2, bl1, Wl2, bl2, out);
}